// BSplineFunction_5935644803138
// MI455X (gfx1250) — compile-verified
//
#include <hip/hip_runtime.h>

// Uniform cubic B-spline basis expansion (1024x8192 -> x8 basis), fp32.
// Bandwidth-bound: 4B in + 32B out per element => ~302MB => ~13us @ 23.3TB/s.
// Closed-form uniform-knot weights (~20 VALU/elem), branchless LDS scatter
// (per-lane indexed ds_store), b128 NT global traffic, guard-free fast path.

typedef float v4f __attribute__((ext_vector_type(4)));

#define TPB   256                      // 8 wave32 per block
#define EPT   4                        // elements per thread
#define EPB   (TPB * EPT)              // 1024 elements per block tile
#define NB    8                        // basis functions per element
#define CHUNK (EPB * NB)               // 8192 floats staged per tile
// pad 4 words every 32 data words: keeps ds b128 16B-aligned, spreads banks,
// and provides one spare ("trash") word per element row for branchless scatter
#define LDSW  (CHUNK + (CHUNK >> 5) * 4)   // 9216 words = 36 KB

__device__ __forceinline__ int pad_idx(int g) { return g + ((g >> 5) << 2); }

struct W4 { float w0, w1, w2, w3; int j0; bool inb; };

__device__ __forceinline__ W4 bspline_w(float t, float k0, float k11, float inv_h)
{
    W4 r;
    r.inb = (t >= k0) && (t < k11);            // level-0 domain (false for NaN)
    const float sf = (t - k0) * inv_h;
    const float fi = __builtin_floorf(sf);
    const float u  = sf - fi;                  // in [0,1)
    r.j0 = (int)fi - 3;                        // leftmost nonzero basis index
    const float c  = 1.0f - u;
    const float u2 = u * u;
    const float u3 = u2 * u;
    const float c3 = c * c * c;
    r.w0 = c3 * (1.0f / 6.0f);
    r.w3 = u3 * (1.0f / 6.0f);
    r.w1 = __builtin_fmaf(0.5f, u3, __builtin_fmaf(-1.0f, u2, 2.0f / 3.0f));
    r.w2 = ((1.0f - r.w0) - r.w1) - r.w3;      // partition of unity
    return r;
}

__global__ __launch_bounds__(TPB)
void bspline8_kernel(const float* __restrict__ x,
                     const float* __restrict__ knots,
                     float* __restrict__ out,
                     long long n)
{
    __shared__ float sm[LDSW];
    const int tid = threadIdx.x;
    const long long eblk = (long long)blockIdx.x * EPB;

    // wave-uniform knot parameters (scalar loads)
    const float k0    = knots[0];
    const float k11   = knots[11];
    const float inv_h = 11.0f / (k11 - k0);

    const v4f z = (v4f)0.0f;
    const bool full = (eblk + EPB) <= n;       // block-uniform branch

    if (full) {
        // ---------------- fast path: zero guards ----------------
        const v4f xv = __builtin_nontemporal_load((const v4f*)(x + eblk) + tid);

        #pragma unroll
        for (int q = 0; q < EPT; ++q) {
            const int r       = tid * EPT + q;
            const int grp     = r >> 2;
            const int rowbase = r * NB + (grp << 2);   // == pad_idx(r*8)
            const int trash   = grp * 36 + 32 + (r & 3); // spare pad word
            *(v4f*)&sm[rowbase]     = z;
            *(v4f*)&sm[rowbase + 4] = z;

            const W4 w = bspline_w(xv[q], k0, k11, inv_h);
            // branchless scatter: out-of-range/out-of-domain -> trash slot
            const int i0 = (w.inb && (unsigned)(w.j0    ) < 8u) ? rowbase + w.j0     : trash;
            const int i1 = (w.inb && (unsigned)(w.j0 + 1) < 8u) ? rowbase + w.j0 + 1 : trash;
            const int i2 = (w.inb && (unsigned)(w.j0 + 2) < 8u) ? rowbase + w.j0 + 2 : trash;
            const int i3 = (w.inb && (unsigned)(w.j0 + 3) < 8u) ? rowbase + w.j0 + 3 : trash;
            sm[i0] = w.w0;
            sm[i1] = w.w1;
            sm[i2] = w.w2;
            sm[i3] = w.w3;
        }

        __syncthreads();

        float* const op = out + eblk * NB;
        #pragma unroll
        for (int s = 0; s < CHUNK / (TPB * 4); ++s) {  // 8 x 4KB coalesced NT
            const int g = (tid + s * TPB) * 4;
            const v4f v = *(const v4f*)&sm[pad_idx(g)];
            __builtin_nontemporal_store(v, (v4f*)(op + g));
        }
    } else {
        // ---------------- guarded tail path (rare) ----------------
        const long long e0 = eblk + (long long)tid * EPT;
        v4f xv = (v4f)(k11 + 1.0f);                    // out of domain -> zeros
        #pragma unroll
        for (int q = 0; q < EPT; ++q)
            if (e0 + q < n) xv[q] = x[e0 + q];

        #pragma unroll
        for (int q = 0; q < EPT; ++q) {
            const int r       = tid * EPT + q;
            const int grp     = r >> 2;
            const int rowbase = r * NB + (grp << 2);
            const int trash   = grp * 36 + 32 + (r & 3);
            *(v4f*)&sm[rowbase]     = z;
            *(v4f*)&sm[rowbase + 4] = z;

            const W4 w = bspline_w(xv[q], k0, k11, inv_h);
            const int i0 = (w.inb && (unsigned)(w.j0    ) < 8u) ? rowbase + w.j0     : trash;
            const int i1 = (w.inb && (unsigned)(w.j0 + 1) < 8u) ? rowbase + w.j0 + 1 : trash;
            const int i2 = (w.inb && (unsigned)(w.j0 + 2) < 8u) ? rowbase + w.j0 + 2 : trash;
            const int i3 = (w.inb && (unsigned)(w.j0 + 3) < 8u) ? rowbase + w.j0 + 3 : trash;
            sm[i0] = w.w0;
            sm[i1] = w.w1;
            sm[i2] = w.w2;
            sm[i3] = w.w3;
        }

        __syncthreads();

        const long long ob  = eblk * NB;
        const long long osz = n * NB;
        #pragma unroll
        for (int s = 0; s < CHUNK / (TPB * 4); ++s) {
            const int g = (tid + s * TPB) * 4;
            const v4f v = *(const v4f*)&sm[pad_idx(g)];
            const long long o = ob + g;
            if (o + 3 < osz) {
                __builtin_nontemporal_store(v, (v4f*)(out + o));
            } else {
                #pragma unroll
                for (int p = 0; p < 4; ++p)
                    if (o + p < osz) out[o + p] = v[p];
            }
        }
    }
}

extern "C" void kernel_launch(void* const* d_in, const int* in_sizes, int n_in,
                              void* d_out, int out_size, void* d_ws, size_t ws_size,
                              hipStream_t stream) {
    const float* x     = (const float*)d_in[0];
    const float* knots = (const float*)d_in[1];
    float*       out   = (float*)d_out;
    const long long n  = in_sizes[0];                  // 1024*8192 elements
    const int nblocks  = (int)((n + EPB - 1) / EPB);   // one 1024-elem tile/block
    bspline8_kernel<<<nblocks, TPB, 0, stream>>>(x, knots, out, n);
}